// TaylorExp_21208548507757
// MI455X (gfx1250) — compile-verified
//
#include <hip/hip_runtime.h>

typedef __attribute__((ext_vector_type(2))) float v2f;
typedef __attribute__((ext_vector_type(8))) float v8f;

// x: [B,H,S,16] flat -> tokens of 16 floats
// out: [B,H,S,273] flat = [1 | x/RRD | (x xT)/(R2*RD)]
// d = 16, RD = 4, RRD = 2, R2 = sqrt(2)
__global__ void taylor_exp_wmma_kernel(const float* __restrict__ x,
                                       float* __restrict__ out,
                                       int ntok) {
    const int lane        = threadIdx.x & 31;
    const int waveInBlock = threadIdx.x >> 5;
    const int wavesPerBlk = blockDim.x >> 5;
    const int wave        = blockIdx.x * wavesPerBlk + waveInBlock;
    const int totalWaves  = gridDim.x * wavesPerBlk;

    const float s_outer = 0.17677669529663687f; // 1/(sqrt(2)*4), folded into A
    const float s_lin   = 0.5f;                 // 1/RRD

    const int  col   = lane & 15;
    const int  baseM = (lane >> 4) << 3;        // 0 for lanes 0-15, 8 for 16-31
    const bool lo    = (lane < 16);
    const bool isL16 = (lane == 16);

    // Branchless term1/term2 slot: lane16 -> offset 0 (writes 1.0);
    // all other lanes -> offset 1+col (lanes 17-31 duplicate lanes 1-15: same
    // value to same address within the wave, benign).
    const int linOff = isL16 ? 0 : (1 + col);

    for (int t = wave; t < ntok; t += totalWaves) {
        const float* xp = x   + (size_t)t * 16;
        float*       op = out + (size_t)t * 273;

        // lanes 0-15 load x[0..15] (one 64B segment); lanes 16-31 reload same
        float xv = xp[col];

        // A: 16x4, only column K=0 nonzero = x * s_outer
        // B: 4x16, only row    K=0 nonzero = x
        v2f a, b;
        a.x = lo ? xv * s_outer : 0.0f;
        a.y = 0.0f;
        b.x = lo ? xv : 0.0f;
        b.y = 0.0f;

        v8f c = {};
        // D = A x B + 0  ->  scaled outer product, 16x16 f32 across the wave
        v8f dmat = __builtin_amdgcn_wmma_f32_16x16x4_f32(
            /*neg_a=*/false, a, /*neg_b=*/false, b,
            /*c_mod=*/(short)0, c, /*reuse_a=*/false, /*reuse_b=*/false);

        // term1 + term2, single unconditional NT store (no exec-mask branches)
        float linVal = isL16 ? 1.0f : xv * s_lin;
        __builtin_nontemporal_store(linVal, op + linOff);

        // term3 (offsets 17..272): VGPR r holds row M=baseM+r, col=lane&15.
        // Each store = two contiguous 64B segments; all 8 form one clause.
        float* qp = op + 17 + col;
        #pragma unroll
        for (int r = 0; r < 8; ++r) {
            __builtin_nontemporal_store(dmat[r], qp + (baseM + r) * 16);
        }
    }
}

extern "C" void kernel_launch(void* const* d_in, const int* in_sizes, int n_in,
                              void* d_out, int out_size, void* d_ws, size_t ws_size,
                              hipStream_t stream) {
    (void)n_in; (void)out_size; (void)d_ws; (void)ws_size;
    const float* x = (const float*)d_in[0];
    float* out = (float*)d_out;

    const int ntok = in_sizes[0] / 16;          // B*H*S tokens of 16 floats

    const int threads = 256;                    // 8 waves per block (wave32)
    const int wavesPerBlk = threads / 32;
    int blocks = (ntok + wavesPerBlk - 1) / wavesPerBlk;
    if (blocks > 65535 * 16) blocks = 65535 * 16; // grid-stride loop covers rest

    taylor_exp_wmma_kernel<<<blocks, threads, 0, stream>>>(x, out, ntok);
}